// DynamicKLDiscretLoss_74749610820261
// MI455X (gfx1250) — compile-verified
//
#include <hip/hip_runtime.h>
#include <math.h>

typedef __attribute__((ext_vector_type(2))) float v2f;
typedef __attribute__((ext_vector_type(8))) float v8f;

#define NEG_INF (-3.0e38f)

__device__ __forceinline__ float wsum(float v) {
#pragma unroll
  for (int m = 16; m > 0; m >>= 1) v += __shfl_xor(v, m, 32);
  return v;
}
__device__ __forceinline__ float wmaxr(float v) {
#pragma unroll
  for (int m = 16; m > 0; m >>= 1) v = fmaxf(v, __shfl_xor(v, m, 32));
  return v;
}

// One workgroup (256 threads = 8 wave32) handles 16 rows of one axis.
// W: bins (384 or 512), KTOP = W/4, F = KTOP+1, FPAD = F rounded up to mult of 4, H = W/8.
template <int W, int KTOP, int F, int FPAD, int H>
__global__ __launch_bounds__(256) void axis_loss_kernel(
    const float* __restrict__ pred, const float* __restrict__ gt,
    const float* __restrict__ tw, const float* __restrict__ w1,
    const float* __restrict__ b1, const float* __restrict__ w2,
    const float* __restrict__ b2, float* __restrict__ out, float inv_joints) {
  constexpr int NELEM = 16;      // 512 virtual sort slots (32 lanes x 16 regs)
  constexpr int NREAL = W / 32;  // real elements per lane
  constexpr int NCH = NREAL / 4; // float4 chunks per lane
  constexpr int KC = FPAD / 4;   // WMMA K-chunks
  constexpr int NT = H / 16;     // WMMA N-tiles

  __shared__ float feats[32][FPAD]; // rows 0-15: pred feats, 16-31: gt feats
  __shared__ float hbuf[32][H];
  __shared__ float beta_lds[32];

  const int tid = threadIdx.x;
  const int lane = tid & 31;
  const int wave = tid >> 5;
  const long long rowBase = (long long)blockIdx.x * 16;

  // ---------- Phase 1: topk (full descending bitonic sort) + mean -> feats ----------
  for (int s = wave * 4; s < wave * 4 + 4; ++s) {
    const float* src = (s < 16) ? pred : gt;
    const int r = (s < 16) ? s : (s - 16);
    const float* row = src + (rowBase + r) * (long long)W;

    float a[NELEM];
    float sum = 0.f;
#pragma unroll
    for (int c = 0; c < NCH; ++c) {
      float4 v4 = *(const float4*)(row + c * 128 + lane * 4);
      a[c * 4 + 0] = v4.x; a[c * 4 + 1] = v4.y;
      a[c * 4 + 2] = v4.z; a[c * 4 + 3] = v4.w;
      sum += v4.x + v4.y + v4.z + v4.w;
    }
#pragma unroll
    for (int i = NREAL; i < NELEM; ++i) a[i] = NEG_INF;
    const float mean = wsum(sum) * (1.0f / (float)W);

    // descending bitonic sort over virtual index v = (i<<5)|lane, N=512
#pragma unroll
    for (int k = 2; k <= 512; k <<= 1) {
#pragma unroll
      for (int j = k >> 1; j > 0; j >>= 1) {
        if (j >= 32) {
          const int jj = j >> 5; // register-index xor
#pragma unroll
          for (int i = 0; i < NELEM; ++i) {
            if ((i & jj) == 0) {
              const int p = i | jj;
              const bool up = (((i << 5) & k) == 0);
              float lo = a[i], hi = a[p];
              const bool sw = up ? (lo < hi) : (lo > hi);
              if (sw) { a[i] = hi; a[p] = lo; }
            }
          }
        } else {
#pragma unroll
          for (int i = 0; i < NELEM; ++i) {
            float o = __shfl_xor(a[i], j, 32);
            const int v = (i << 5) | lane;
            const bool up = ((v & k) == 0);
            const bool low = ((lane & j) == 0);
            float mx = fmaxf(a[i], o), mn = fminf(a[i], o);
            a[i] = (up == low) ? mx : mn;
          }
        }
      }
    }

    // rank v holds v-th largest; take sorted top-KTOP, append mean, zero-pad
#pragma unroll
    for (int i = 0; i < NELEM; ++i) {
      const int v = (i << 5) | lane;
      if (v < KTOP) feats[s][v] = a[i];
    }
    if (lane == 0) feats[s][KTOP] = mean;
    for (int f = F + lane; f < FPAD; f += 32) feats[s][f] = 0.f;
  }
  __syncthreads();

  // ---------- Phase 2a: MLP layer 1 via V_WMMA_F32_16X16X4_F32 ----------
  // wave 0 -> pred feats (rows 0-15), wave 1 -> gt feats (rows 16-31)
  if (wave < 2) {
    const int base = wave * 16;
    const int m = lane & 15;    // A: row M ; B/C: col N
    const int half = lane >> 4; // K sub-pair select
#pragma unroll
    for (int nt = 0; nt < NT; ++nt) {
      v8f c = 0.f;
      for (int kc = 0; kc < KC; ++kc) {
        const int k0 = kc * 4 + half * 2;
        v2f av, bv;
        av.x = feats[base + m][k0];
        av.y = feats[base + m][k0 + 1];
        bv.x = (k0 < F) ? w1[k0 * H + nt * 16 + m] : 0.f;
        bv.y = (k0 + 1 < F) ? w1[(k0 + 1) * H + nt * 16 + m] : 0.f;
        c = __builtin_amdgcn_wmma_f32_16x16x4_f32(
            false, av, false, bv, (short)0, c, false, false);
      }
#pragma unroll
      for (int r8 = 0; r8 < 8; ++r8) {
        const int rowm = half * 8 + r8; // C layout: VGPR r -> M=r (lanes 0-15) / M=8+r
        const float hv = c[r8] + b1[nt * 16 + m];
        hbuf[base + rowm][nt * 16 + m] = fmaxf(hv, 0.f);
      }
    }
  }
  __syncthreads();

  // ---------- Phase 2b: MLP layer 2 -> beta = 1 + sigmoid(h.w2 + b2) ----------
  if (tid < 32) {
    float z = b2[0];
#pragma unroll
    for (int j = 0; j < H; ++j) z += hbuf[tid][j] * w2[j];
    beta_lds[tid] = 1.0f + 1.0f / (1.0f + __expf(-z));
  }
  __syncthreads();

  // ---------- Phase 3: KL(labels || scores), weighted accumulate ----------
  float acc = 0.f;
#pragma unroll
  for (int rr = 0; rr < 2; ++rr) {
    const int r = wave * 2 + rr;
    const float bp = beta_lds[r];
    const float bg = beta_lds[16 + r];
    const float* prow = pred + (rowBase + r) * (long long)W;
    const float* grow = gt + (rowBase + r) * (long long)W;

    float sp[NREAL], sg[NREAL];
    float mp = NEG_INF, mg = NEG_INF;
#pragma unroll
    for (int c = 0; c < NCH; ++c) {
      float4 p4 = *(const float4*)(prow + c * 128 + lane * 4);
      float4 g4 = *(const float4*)(grow + c * 128 + lane * 4);
      sp[c * 4 + 0] = p4.x * bp; sp[c * 4 + 1] = p4.y * bp;
      sp[c * 4 + 2] = p4.z * bp; sp[c * 4 + 3] = p4.w * bp;
      sg[c * 4 + 0] = g4.x * bg; sg[c * 4 + 1] = g4.y * bg;
      sg[c * 4 + 2] = g4.z * bg; sg[c * 4 + 3] = g4.w * bg;
    }
#pragma unroll
    for (int i = 0; i < NREAL; ++i) {
      mp = fmaxf(mp, sp[i]);
      mg = fmaxf(mg, sg[i]);
    }
    mp = wmaxr(mp); mg = wmaxr(mg);
    float sump = 0.f, sumg = 0.f;
#pragma unroll
    for (int i = 0; i < NREAL; ++i) {
      sump += __expf(sp[i] - mp);
      sumg += __expf(sg[i] - mg);
    }
    sump = wsum(sump); sumg = wsum(sumg);
    const float lsep = mp + __logf(sump);
    const float lseg = mg + __logf(sumg);
    float t = 0.f;
#pragma unroll
    for (int i = 0; i < NREAL; ++i) {
      const float ll = sg[i] - lseg;                 // log_labels
      t += __expf(ll) * (ll - (sp[i] - lsep));       // labels*(log_labels - scores)
    }
    t = wsum(t);
    if (lane == 0) acc += (t * (1.0f / (float)W)) * tw[rowBase + r];
  }
  if (lane == 0) atomicAdd(out, acc * inv_joints);
}

__global__ void zero_out_kernel(float* o) { o[0] = 0.f; }

extern "C" void kernel_launch(void* const* d_in, const int* in_sizes, int n_in,
                              void* d_out, int out_size, void* d_ws, size_t ws_size,
                              hipStream_t stream) {
  const float* output_x = (const float*)d_in[0];
  const float* output_y = (const float*)d_in[1];
  const float* target_x = (const float*)d_in[2];
  const float* target_y = (const float*)d_in[3];
  const float* tweight  = (const float*)d_in[4];
  const float* fcx_w1 = (const float*)d_in[5];
  const float* fcx_b1 = (const float*)d_in[6];
  const float* fcx_w2 = (const float*)d_in[7];
  const float* fcx_b2 = (const float*)d_in[8];
  const float* fcy_w1 = (const float*)d_in[9];
  const float* fcy_b1 = (const float*)d_in[10];
  const float* fcy_w2 = (const float*)d_in[11];
  const float* fcy_b2 = (const float*)d_in[12];
  float* out = (float*)d_out;

  const int rowsBK = in_sizes[4];      // B*K = 34816
  const int nblk = rowsBK / 16;        // 2176
  const float inv_joints = 1.0f / 17.0f;

  zero_out_kernel<<<1, 1, 0, stream>>>(out);
  axis_loss_kernel<384, 96, 97, 100, 48><<<nblk, 256, 0, stream>>>(
      output_x, target_x, tweight, fcx_w1, fcx_b1, fcx_w2, fcx_b2, out, inv_joints);
  axis_loss_kernel<512, 128, 129, 132, 64><<<nblk, 256, 0, stream>>>(
      output_y, target_y, tweight, fcy_w1, fcy_b1, fcy_w2, fcy_b2, out, inv_joints);
}